// BigramModel_23828478558312
// MI455X (gfx1250) — compile-verified
//
#include <hip/hip_runtime.h>

// ---- problem constants (match reference) ----
constexpr int NV = 4096;   // vocab
constexpr int NS = 256;    // sequence
constexpr int NB = 64;     // batch
constexpr float F_ALPHA = 0.4f;                      // bigram weight
constexpr float F_BETA  = 0.3f;                      // trigram weight
constexpr float F_UNI   = (float)(1.0 - 0.4 - 0.3);  // == 0.3f (matches ref f64->f32)
constexpr float F_EPS   = 1e-10f;

typedef float v4f __attribute__((ext_vector_type(4)));

// One 256-thread block (8 wave32) per (s,b) token position.
// LDS: [0,16K) bigram row, [16K,32K) trigram row (only when needed), + 8 f32 red slots.
__global__ __launch_bounds__(256) void ngram_logprob_kernel(
    const int*   __restrict__ text,      // [NS, NB]
    const float* __restrict__ unigram,   // [NV]
    const float* __restrict__ bigram,    // [NV, NV]
    const float* __restrict__ tri_rows,  // [K, NV]
    const int*   __restrict__ tri_map,   // [NV*NV]
    float*       __restrict__ out)       // [NS, NB, NV]
{
    __shared__ __align__(16) float s_rows[2 * NV];
    __shared__ float s_red[8];

    const int tid  = threadIdx.x;
    const int blk  = blockIdx.x;       // blk = s*NB + b
    const int sPos = blk >> 6;         // / NB (NB == 64)

    // ---- uniform per-block context (scalarized s_load's) ----
    const int w2 = text[blk];
    const int w1 = (sPos > 0) ? text[blk - NB] : text[blk & (NB - 1)]; // prev[0] = text[0]
    const int rowIdx = tri_map[w1 * NV + w2];
    // Trigram contributes only when the (w1,w2) context was observed AND s > 1.
    // With K=20000 of 16.7M keys live this is true for ~0.1% of blocks.
    const bool useTri = (rowIdx >= 0) && (sPos > 1);

    const unsigned ldsBase =
        (unsigned)(unsigned long long)(uintptr_t)(void*)s_rows;   // low 32 bits = LDS offset
    const unsigned long long gBi =
        (unsigned long long)(uintptr_t)(bigram + (size_t)w2 * NV);

    // ---- CDNA5 async global -> LDS staging (each lane moves 16B per op) ----
#pragma unroll
    for (int c = 0; c < 4; ++c) {
        const unsigned off = (unsigned)(tid * 16 + c * 4096);
        const unsigned ldsBi = ldsBase + off;
        const unsigned long long aBi = gBi + off;
        asm volatile("global_load_async_to_lds_b128 %0, %1, off"
                     :: "v"(ldsBi), "v"(aBi) : "memory");
    }
    if (useTri) {   // block-uniform branch: EXEC stays all-ones inside
        const unsigned long long gTri =
            (unsigned long long)(uintptr_t)(tri_rows + (size_t)rowIdx * NV);
#pragma unroll
        for (int c = 0; c < 4; ++c) {
            const unsigned off = (unsigned)(tid * 16 + c * 4096);
            const unsigned ldsTri = ldsBase + 16384u + off;
            const unsigned long long aTri = gTri + off;
            asm volatile("global_load_async_to_lds_b128 %0, %1, off"
                         :: "v"(ldsTri), "v"(aTri) : "memory");
        }
    }

    // ---- overlap: fetch unigram into registers while the DMA fills LDS ----
    const v4f* uni4 = (const v4f*)unigram;
    v4f u[4];
#pragma unroll
    for (int c = 0; c < 4; ++c) u[c] = uni4[tid + c * 256];

    asm volatile("s_wait_asynccnt 0x0" ::: "memory");
    __syncthreads();

    // ---- blend + partial row sum; keep the 16 probs in registers ----
    v4f p[4];
    float lsum = 0.0f;
    if (useTri) {
#pragma unroll
        for (int c = 0; c < 4; ++c) {
            const int e4 = tid + c * 256;
            const v4f bi = *(const v4f*)&s_rows[e4 * 4];        // ds_load_b128
            const v4f tr = *(const v4f*)&s_rows[NV + e4 * 4];
            const v4f pr = F_UNI * u[c] + F_ALPHA * bi + F_BETA * tr;
            p[c] = pr;
            lsum += (pr.x + pr.y) + (pr.z + pr.w);
        }
    } else {
#pragma unroll
        for (int c = 0; c < 4; ++c) {
            const int e4 = tid + c * 256;
            const v4f bi = *(const v4f*)&s_rows[e4 * 4];
            const v4f pr = F_UNI * u[c] + F_ALPHA * bi;
            p[c] = pr;
            lsum += (pr.x + pr.y) + (pr.z + pr.w);
        }
    }

    // ---- block reduction over V: wave32 shfl tree, then 8 wave partials ----
#pragma unroll
    for (int m = 16; m >= 1; m >>= 1)
        lsum += __shfl_xor(lsum, m, 32);
    if ((tid & 31) == 0) s_red[tid >> 5] = lsum;
    __syncthreads();
    float den = F_EPS;
#pragma unroll
    for (int w = 0; w < 8; ++w) den += s_red[w];
    const float rden = 1.0f / den;

    // ---- normalize, log, streaming (non-temporal) b128 stores ----
    v4f* out4 = (v4f*)(out + (size_t)blk * NV);
#pragma unroll
    for (int c = 0; c < 4; ++c) {
        const int e4 = tid + c * 256;
        const v4f pn = p[c] * rden;
        v4f lg;
        lg.x = __logf(F_EPS + pn.x);
        lg.y = __logf(F_EPS + pn.y);
        lg.z = __logf(F_EPS + pn.z);
        lg.w = __logf(F_EPS + pn.w);
        __builtin_nontemporal_store(lg, &out4[e4]);   // keep bigram table hot in L2
    }
}

extern "C" void kernel_launch(void* const* d_in, const int* in_sizes, int n_in,
                              void* d_out, int out_size, void* d_ws, size_t ws_size,
                              hipStream_t stream) {
    const int*   text     = (const int*)  d_in[0];
    const float* unigram  = (const float*)d_in[1];
    const float* bigram   = (const float*)d_in[2];
    const float* tri_rows = (const float*)d_in[3];
    const int*   tri_map  = (const int*)  d_in[4];
    float* out = (float*)d_out;

    dim3 grid(NS * NB);   // 16384 blocks, one per (s, b)
    dim3 block(256);      // 8 wave32 waves
    ngram_logprob_kernel<<<grid, block, 0, stream>>>(
        text, unigram, bigram, tri_rows, tri_map, out);
}